// CustomRandomEqualize_24094766530585
// MI455X (gfx1250) — compile-verified
//
#include <hip/hip_runtime.h>

typedef float v2f __attribute__((ext_vector_type(2)));
typedef float v8f __attribute__((ext_vector_type(8)));

#define NUM_LABELS 3
#define IMG_H 2048
#define IMG_W 4096
#define CH_ELEMS (IMG_H * IMG_W)   // 8,388,608 per channel
#define NUM_BINS 256

// ---------------------------------------------------------------------------
// Zero the per-channel histograms in workspace (d_ws is poisoned to 0xAA).
// ---------------------------------------------------------------------------
__global__ void zero_hist_kernel(int* __restrict__ hist) {
    for (int i = threadIdx.x; i < 3 * NUM_BINS; i += blockDim.x) hist[i] = 0;
}

// ---------------------------------------------------------------------------
// Per-channel 256-bin histogram. blockIdx.y = channel. 8 per-wave LDS
// sub-histograms per block (no inter-wave LDS contention), one global
// atomic per bin per block at the end. float4 (B128) streaming loads.
// ---------------------------------------------------------------------------
__global__ __launch_bounds__(256) void hist_kernel(const float* __restrict__ img,
                                                   int* __restrict__ hist,
                                                   int blocks_x) {
    __shared__ unsigned int whist[8 * NUM_BINS];
    const int ch  = blockIdx.y;
    const int t   = threadIdx.x;
    const int wid = t >> 5;   // wave32: wave id within block

    for (int i = t; i < 8 * NUM_BINS; i += 256) whist[i] = 0u;
    __syncthreads();

    const float4* __restrict__ src =
        reinterpret_cast<const float4*>(img + (size_t)ch * CH_ELEMS);
    const int n4     = CH_ELEMS / 4;
    const int stride = blocks_x * 256;
    unsigned int* mh = &whist[wid * NUM_BINS];

    for (int i = blockIdx.x * 256 + t; i < n4; i += stride) {
        int ip = i + stride;
        if (ip < n4) __builtin_prefetch(&src[ip], 0, 0);  // global_prefetch_b8
        float4 v = src[i];
        atomicAdd(&mh[(int)v.x], 1u);
        atomicAdd(&mh[(int)v.y], 1u);
        atomicAdd(&mh[(int)v.z], 1u);
        atomicAdd(&mh[(int)v.w], 1u);
    }
    __syncthreads();

    unsigned int s = 0;
    for (int w = 0; w < 8; ++w) s += whist[w * NUM_BINS + t];
    if (s) atomicAdd((unsigned int*)&hist[ch * NUM_BINS + t], s);
}

// ---------------------------------------------------------------------------
// LUT builder: one wave (32 threads) per channel.
// cumsum(hist) via WMMA: M[r][c] = hist[16r+c]; S = M x U with U = upper-
// triangular ones, computed as 4 chained V_WMMA_F32_16X16X4_F32 (K slices).
// Counts <= 2^23 so every partial sum is exact in f32.
// Operand layouts (ISA 7.12.2, 32-bit, wave32):
//   A 16x4: lane -> row (lane&15), K = vgpr + 2*(lane>>4)
//   B 4x16: lane -> col (lane&15), K = vgpr + 2*(lane>>4)
//   C/D 16x16: vgpr v -> row v + 8*(lane>>4), col = lane&15
// Row offsets (16-element exclusive scan) + step/last_nz done serially.
// ---------------------------------------------------------------------------
__global__ __launch_bounds__(32) void lut_kernel(const int* __restrict__ hist,
                                                 int* __restrict__ lut) {
    __shared__ int sCum[NUM_BINS];
    __shared__ int sRowOff[16];
    __shared__ int sStep;

    const int ch   = blockIdx.x;
    const int lane = threadIdx.x;
    const int half = lane >> 4;
    const int idx  = lane & 15;
    const int* __restrict__ h = hist + ch * NUM_BINS;

    v8f acc = {};
#pragma unroll
    for (int q = 0; q < 4; ++q) {
        v2f a, b;
#pragma unroll
        for (int j = 0; j < 2; ++j) {
            const int k = 4 * q + 2 * half + j;          // K index of this slice
            a[j] = (float)h[idx * 16 + k];               // A = M, row=idx
            b[j] = (k <= idx) ? 1.0f : 0.0f;             // B = U, col=idx
        }
        // D = A*B + C  (accumulator chained across the 4 K-slices)
        acc = __builtin_amdgcn_wmma_f32_16x16x4_f32(
            /*neg_a=*/false, a, /*neg_b=*/false, b,
            /*c_mod=*/(short)0, acc, /*reuse_a=*/false, /*reuse_b=*/false);
    }

    // Scatter S[r][c] (inclusive row scans) into LDS as exact integers.
#pragma unroll
    for (int v = 0; v < 8; ++v) {
        const int r = v + 8 * half;
        sCum[r * 16 + idx] = __float2int_rn(acc[v]);
    }
    __syncthreads();

    if (lane == 0) {
        int off = 0;
        for (int r = 0; r < 16; ++r) { sRowOff[r] = off; off += sCum[r * 16 + 15]; }
        const int total = off;
        int last_nz = 0;
        for (int b = NUM_BINS - 1; b >= 0; --b) { if (h[b] > 0) { last_nz = b; break; } }
        sStep = (total - h[last_nz]) / (NUM_BINS - 1);
    }
    __syncthreads();

    const int step = sStep;
#pragma unroll
    for (int j = 0; j < 8; ++j) {
        const int i = lane * 8 + j;
        sCum[i] += sRowOff[i >> 4];          // full cumulative sum cum[i]
    }
    __syncthreads();

    const int smax = step > 0 ? step : 1;
#pragma unroll
    for (int j = 0; j < 8; ++j) {
        const int b = lane * 8 + j;
        int v;
        if (step == 0)      v = b;           // identity (reference still truncates)
        else if (b == 0)    v = 0;
        else {
            v = (sCum[b - 1] + step / 2) / smax;
            v = v < 0 ? 0 : (v > 255 ? 255 : v);
        }
        lut[ch * NUM_BINS + b] = v;
    }
}

// ---------------------------------------------------------------------------
// Apply pass: channels 0..2 gather through the LUT (LDS-resident),
// channels 3..5 are a straight float4 copy. blockIdx.y = channel.
// ---------------------------------------------------------------------------
__global__ __launch_bounds__(256) void apply_kernel(const float* __restrict__ img,
                                                    const int* __restrict__ lut,
                                                    float* __restrict__ out,
                                                    int blocks_x) {
    __shared__ float sLut[NUM_BINS];
    const int ch = blockIdx.y;
    const int t  = threadIdx.x;
    const bool eq = (ch < 3);
    if (eq) {
        sLut[t] = (float)lut[ch * NUM_BINS + t];   // blockDim==256==NUM_BINS
        __syncthreads();
    }

    const size_t base = (size_t)ch * CH_ELEMS;
    const float4* __restrict__ src = reinterpret_cast<const float4*>(img + base);
    float4* __restrict__ dst       = reinterpret_cast<float4*>(out + base);
    const int n4     = CH_ELEMS / 4;
    const int stride = blocks_x * 256;

    for (int i = blockIdx.x * 256 + t; i < n4; i += stride) {
        int ip = i + stride;
        if (ip < n4) __builtin_prefetch(&src[ip], 0, 0);
        float4 v = src[i];
        if (eq) {
            v.x = sLut[(int)v.x];
            v.y = sLut[(int)v.y];
            v.z = sLut[(int)v.z];
            v.w = sLut[(int)v.w];
        }
        dst[i] = v;
    }
}

// ---------------------------------------------------------------------------
extern "C" void kernel_launch(void* const* d_in, const int* in_sizes, int n_in,
                              void* d_out, int out_size, void* d_ws, size_t ws_size,
                              hipStream_t stream) {
    (void)in_sizes; (void)n_in; (void)out_size; (void)ws_size;
    const float* img = (const float*)d_in[0];      // [6, 2048, 4096] f32
    float* out       = (float*)d_out;              // [6, 2048, 4096] f32
    int* hist        = (int*)d_ws;                 // 3*256 ints
    int* lut         = hist + 3 * NUM_BINS;        // 3*256 ints

    const int histBlocksX  = 512;
    const int applyBlocksX = 1024;

    zero_hist_kernel<<<1, 256, 0, stream>>>(hist);
    hist_kernel<<<dim3(histBlocksX, 3), 256, 0, stream>>>(img, hist, histBlocksX);
    lut_kernel<<<3, 32, 0, stream>>>(hist, lut);
    apply_kernel<<<dim3(applyBlocksX, 3 + NUM_LABELS), 256, 0, stream>>>(img, lut, out, applyBlocksX);
}